// AugmentPipe_34248069218753
// MI455X (gfx1250) — compile-verified
//
#include <hip/hip_runtime.h>
#include <math.h>

// ---------------------------------------------------------------------------
// AugmentPipe: flip -> affine warp (bilinear + reflect) -> bright/contrast/sat
//              -> cutout(noise).  B=64, C=3, H=W=512, fp32.
// Memory-bound (~0.6 GB @ 23.3 TB/s => ~25-30us). No matmul => no WMMA; the
// CDNA5-specific wins are NT cache hints for streaming traffic and async
// global->LDS noise prefetch (ASYNCcnt) overlapped with the gather pipeline.
// ---------------------------------------------------------------------------

#define IMG_W   512
#define IMG_H   512
#define HWSZ    (IMG_W * IMG_H)      // 262144 = 1<<18
#define CUT_SZ  256                  // int(512 * 0.5)

#if defined(__has_builtin)
#  if __has_builtin(__builtin_amdgcn_global_load_async_to_lds_b32)
#    define HAVE_ASYNC_LDS 1
#  endif
#  if __has_builtin(__builtin_amdgcn_s_wait_asynccnt)
#    define HAVE_WAIT_ASYNC 1
#  endif
#endif
#ifndef HAVE_ASYNC_LDS
#  define HAVE_ASYNC_LDS 0
#endif
#ifndef HAVE_WAIT_ASYNC
#  define HAVE_WAIT_ASYNC 0
#endif

#if HAVE_ASYNC_LDS
typedef __attribute__((address_space(1))) int g_int_t;   // global
typedef __attribute__((address_space(3))) int l_int_t;   // LDS
#endif

__device__ __forceinline__ float reflect_coord(float v, float size) {
    v = fabsf(v + 0.5f);
    v = fmodf(v, 2.0f * size);
    v = fminf(v, 2.0f * size - v);
    return fminf(fmaxf(v - 0.5f, 0.0f), size - 1.0f);
}

__device__ __forceinline__ float clamp1(float v) {
    return fminf(fmaxf(v, -1.0f), 1.0f);
}

__global__ __launch_bounds__(256)
void augment_kernel(const float* __restrict__ images,
                    const float* __restrict__ u_angle,
                    const float* __restrict__ u_scale,
                    const float* __restrict__ u_trans,
                    const float* __restrict__ u_bright,
                    const float* __restrict__ u_contrast,
                    const float* __restrict__ u_sat,
                    const float* __restrict__ noise,
                    const int*   __restrict__ m_flip,
                    const int*   __restrict__ m_rot,
                    const int*   __restrict__ m_scale,
                    const int*   __restrict__ m_trans,
                    const int*   __restrict__ m_bright,
                    const int*   __restrict__ m_contrast,
                    const int*   __restrict__ m_sat,
                    const int*   __restrict__ m_cut,
                    const int*   __restrict__ y0a,
                    const int*   __restrict__ x0a,
                    float*       __restrict__ out)
{
    const int b   = blockIdx.y;                 // wave-uniform image index
    const int tid = threadIdx.x;
    const int p   = blockIdx.x * 256 + tid;     // pixel index within image
    const int y   = p >> 9;                     // /512
    const int x   = p & 511;                    // %512

    // ---- per-image parameters (uniform -> scalar loads / SALU) ----
    const int   flip = m_flip[b];
    const float ang  = (m_rot[b]      > 0) ? (u_angle[b] * 2.0f - 1.0f) * 3.14159265358979323846f : 0.0f;
    const float sc   = (m_scale[b]    > 0) ? (u_scale[b] * 2.0f - 1.0f) * 0.2f + 1.0f : 1.0f;
    const float tr   = (m_trans[b]    > 0) ? (u_trans[b] * 2.0f - 1.0f) * 0.125f : 0.0f;
    const float bb   = (m_bright[b]   > 0) ? u_bright[b] * 0.2f : 0.0f;
    const float cc   = (m_contrast[b] > 0) ? u_contrast[b] + 0.5f : 1.0f;
    const float ss   = (m_sat[b]      > 0) ? u_sat[b] * 2.0f : 1.0f;
    const int   cutB = m_cut[b];
    const int   cy0  = y0a[b];
    const int   cx0  = x0a[b];

    float sa, ca;
    sincosf(ang, &sa, &ca);

    // ---- affine grid + reflect-bilinear source coords ----
    const float gx0 = (float)x * (2.0f / 511.0f) - 1.0f;
    const float gy0 = (float)y * (2.0f / 511.0f) - 1.0f;
    const float gx  = sc * (ca * gx0 - sa * gy0) + tr;
    const float gy  = sc * (sa * gx0 + ca * gy0) + tr;

    float xs = ((gx + 1.0f) * (float)IMG_W - 1.0f) * 0.5f;
    float ys = ((gy + 1.0f) * (float)IMG_H - 1.0f) * 0.5f;
    xs = reflect_coord(xs, (float)IMG_W);
    ys = reflect_coord(ys, (float)IMG_H);

    const float xf = floorf(xs), yf = floorf(ys);
    const float wx = xs - xf,    wy = ys - yf;
    int x0i = (int)fminf(fmaxf(xf,        0.0f), 511.0f);
    int x1i = (int)fminf(fmaxf(xf + 1.0f, 0.0f), 511.0f);
    int y0i = (int)fminf(fmaxf(yf,        0.0f), 511.0f);
    int y1i = (int)fminf(fmaxf(yf + 1.0f, 0.0f), 511.0f);
    if (flip > 0) { x0i = 511 - x0i; x1i = 511 - x1i; }   // flip == source x remap

    const int i00 = y0i * IMG_W + x0i, i01 = y0i * IMG_W + x1i;
    const int i10 = y1i * IMG_W + x0i, i11 = y1i * IMG_W + x1i;

    // ---- block-uniform cutout predicate; async-prefetch noise into LDS ----
    // Each block covers exactly half of one row => y and x-range are uniform.
    const int  xr0      = (blockIdx.x & 1) << 8;
    const bool blockCut = (cutB > 0) && (y >= cy0) && (y < cy0 + CUT_SZ) &&
                          (cx0 < xr0 + 256) && (cx0 + CUT_SZ > xr0);

    __shared__ float s_noise[3 * 256];
    const size_t nbase = (size_t)b * 3u * HWSZ + (size_t)p;

#if HAVE_ASYNC_LDS
    if (blockCut) {
        #pragma unroll
        for (int c = 0; c < 3; ++c) {
            __builtin_amdgcn_global_load_async_to_lds_b32(
                (g_int_t*)(noise + nbase + (size_t)c * HWSZ),
                (l_int_t*)&s_noise[c * 256 + tid],
                0, 0);
        }
    }
#endif

    // ---- gather + bilinear for all 3 channels (overlaps async noise load) ----
    const float* img = images + (size_t)b * 3u * HWSZ;
    float v[3];
    #pragma unroll
    for (int c = 0; c < 3; ++c) {
        const float* pl = img + (size_t)c * HWSZ;
        const float v00 = pl[i00], v01 = pl[i01];
        const float v10 = pl[i10], v11 = pl[i11];
        const float top = v00 + wx * (v01 - v00);
        const float bot = v10 + wx * (v11 - v10);
        v[c] = top + wy * (bot - top);
    }

    // ---- brightness, contrast, saturation ----
    #pragma unroll
    for (int c = 0; c < 3; ++c) {
        float t = clamp1(v[c] + bb);
        v[c]    = clamp1(t * cc);
    }
    const float gray = (v[0] + v[1] + v[2]) * (1.0f / 3.0f);
    #pragma unroll
    for (int c = 0; c < 3; ++c)
        v[c] = clamp1(gray + ss * (v[c] - gray));

    // ---- cutout: replace with noise ----
    const bool myCut = blockCut && (x >= cx0) && (x < cx0 + CUT_SZ);
#if HAVE_ASYNC_LDS
    if (blockCut) {
#if HAVE_WAIT_ASYNC
        __builtin_amdgcn_s_wait_asynccnt(0);
#else
        asm volatile("s_wait_asynccnt 0" ::: "memory");
#endif
        asm volatile("" ::: "memory");    // fence compiler ordering for LDS reads
    }
    if (myCut) {
        v[0] = s_noise[tid];
        v[1] = s_noise[256 + tid];
        v[2] = s_noise[512 + tid];
    }
#else
    if (myCut) {
        #pragma unroll
        for (int c = 0; c < 3; ++c)
            v[c] = __builtin_nontemporal_load(noise + nbase + (size_t)c * HWSZ);
    }
#endif

    // ---- streaming (non-temporal) store: write-once, never re-read ----
    float* o = out + (size_t)b * 3u * HWSZ + (size_t)p;
    __builtin_nontemporal_store(v[0], o);
    __builtin_nontemporal_store(v[1], o + HWSZ);
    __builtin_nontemporal_store(v[2], o + 2 * HWSZ);
}

extern "C" void kernel_launch(void* const* d_in, const int* in_sizes, int n_in,
                              void* d_out, int out_size, void* d_ws, size_t ws_size,
                              hipStream_t stream) {
    (void)in_sizes; (void)n_in; (void)out_size; (void)d_ws; (void)ws_size;
    const float* images     = (const float*)d_in[0];
    const float* u_angle    = (const float*)d_in[1];
    const float* u_scale    = (const float*)d_in[2];
    const float* u_trans    = (const float*)d_in[3];
    const float* u_bright   = (const float*)d_in[4];
    const float* u_contrast = (const float*)d_in[5];
    const float* u_sat      = (const float*)d_in[6];
    const float* noise      = (const float*)d_in[7];
    const int*   m_flip     = (const int*)d_in[8];
    const int*   m_rot      = (const int*)d_in[9];
    const int*   m_scale    = (const int*)d_in[10];
    const int*   m_trans    = (const int*)d_in[11];
    const int*   m_bright   = (const int*)d_in[12];
    const int*   m_contrast = (const int*)d_in[13];
    const int*   m_sat      = (const int*)d_in[14];
    const int*   m_cut      = (const int*)d_in[15];
    const int*   y0a        = (const int*)d_in[16];
    const int*   x0a        = (const int*)d_in[17];
    float* out = (float*)d_out;

    dim3 grid(HWSZ / 256, 64);   // 1024 half-row blocks per image, 64 images
    dim3 block(256);             // 8 wave32 waves
    augment_kernel<<<grid, block, 0, stream>>>(
        images, u_angle, u_scale, u_trans, u_bright, u_contrast, u_sat, noise,
        m_flip, m_rot, m_scale, m_trans, m_bright, m_contrast, m_sat, m_cut,
        y0a, x0a, out);
}